// VSCoding_70231305224554
// MI455X (gfx1250) — compile-verified
//
#include <hip/hip_runtime.h>
#include <cstddef>
#include <cstdint>

// MI455X / gfx1250, wave32. fp32 GEMM via V_WMMA_F32_16X16X4_F32,
// staged through LDS with GLOBAL_LOAD_ASYNC_TO_LDS_B128 (ASYNCcnt), double-buffered.
typedef float v2f __attribute__((ext_vector_type(2)));
typedef float v8f __attribute__((ext_vector_type(8)));

#define N_IMG 128
#define MAX_R 36
#define N_CAP 128
#define MAX_W 32
#define DDIM  512
#define RPAD  48          // r padded to 3 WMMA tiles
#define NEGV  (-1e30f)
#define SROW  53          // odd LDS row stride for sims/attn -> conflict-free
#define KCH   64          // K-chunk staged per double-buffer slot
#define LSTR  68          // stage row stride (dwords): 68%64==4 -> conflict-free b64 frags,
                          // 272B row stride keeps b128 LDS writes 16B-aligned
#define NCHUNK (DDIM / KCH)

__device__ __forceinline__ void async_b128(const float* gaddr, uint32_t lds_byte) {
    // GLOBAL_LOAD_ASYNC_TO_LDS_B128: VDST = LDS byte address, VADDR = 64-bit global addr.
    asm volatile("global_load_async_to_lds_b128 %0, %1, off"
                 :: "v"(lds_byte), "v"(gaddr) : "memory");
}
__device__ __forceinline__ void wait_async0() {
    asm volatile("s_wait_asynccnt 0" ::: "memory");
}

// ---------------------------------------------------------------------------
// Kernel 1: per-image Gram matrices G[i] = imgs[i] @ imgs[i]^T  (48x48, padded)
// 9 waves per image, one 16x16 WMMA tile each, K = 512. Rows >= 36 clamped
// (those G entries are never read).
// ---------------------------------------------------------------------------
__global__ __launch_bounds__(288) void gram_kernel(const float* __restrict__ imgs,
                                                   float* __restrict__ G) {
    const int i    = blockIdx.x;
    const int wid  = threadIdx.x >> 5;        // 0..8
    const int lane = threadIdx.x & 31;
    const int mt   = wid / 3, nt = wid % 3;
    const int l15  = lane & 15;
    const int kl   = (lane >> 4) * 2;         // lanes 16-31 hold K+2,K+3

    const float* base = imgs + (size_t)i * MAX_R * DDIM;
    const float* pa = base + (size_t)min(mt * 16 + l15, MAX_R - 1) * DDIM + kl;
    const float* pb = base + (size_t)min(nt * 16 + l15, MAX_R - 1) * DDIM + kl;

    v8f acc = {};
#pragma unroll 8
    for (int k = 0; k < DDIM; k += 4) {
        v2f a = *(const v2f*)(pa + k);
        v2f b = *(const v2f*)(pb + k);
        acc = __builtin_amdgcn_wmma_f32_16x16x4_f32(false, a, false, b,
                                                    (short)0, acc, false, false);
    }

    float* g = G + (size_t)i * RPAD * RPAD;
    const int row0 = mt * 16 + ((lane & 16) ? 8 : 0);
    const int col  = nt * 16 + l15;
#pragma unroll
    for (int n = 0; n < 8; ++n)
        g[(size_t)(row0 + n) * RPAD + col] = acc[n];
}

// ---------------------------------------------------------------------------
// Kernel 2: one (c,i) pair per 192-thread block (6 waves).
// Phase 1: async-stage caps(32xKCH) + imgs(48xKCH) chunks into LDS (double
//          buffered), 6 WMMA tiles read fragments via ds_load_b64.
// Phase 2: per-row top-5 threshold + softmax; num = sum_r attn*S_raw.
// Phase 3: den^2 = attn^T G attn via per-image Gram matrix.
// ---------------------------------------------------------------------------
__global__ __launch_bounds__(192) void vsc_main(const float* __restrict__ imgs,
                                                const float* __restrict__ caps,
                                                const int* __restrict__ img_lens,
                                                const int* __restrict__ cap_lens,
                                                const float* __restrict__ G,
                                                float* __restrict__ out) {
    // stage: caps rows 0..31, imgs rows 32..79 ; two buffers for overlap
    __shared__ __align__(16) float s_stage[2][(MAX_W + RPAD) * LSTR];
    __shared__ float s_mat[MAX_W * SROW];   // raw sims, later overwritten with attn
    __shared__ float s_pd[192];             // den^2 partials

    const int c    = blockIdx.x;
    const int i    = blockIdx.y;
    const int t    = threadIdx.x;
    const int wid  = t >> 5;
    const int lane = t & 31;
    const int l15  = lane & 15;
    const int kl   = (lane >> 4) * 2;

    const float* caps_c = caps + ((size_t)c * MAX_W) * DDIM;
    const float* imgs_i = imgs + ((size_t)i * MAX_R) * DDIM;
    const uint32_t base0 = (uint32_t)(uintptr_t)&s_stage[0][0];
    const uint32_t base1 = (uint32_t)(uintptr_t)&s_stage[1][0];

    // ---- async stage of one K-chunk: 512 + 768 b128 transfers over 192 threads
    auto stage_chunk = [&](uint32_t lds_base, int kbase) {
        // caps: 32 rows x 16 b128  (last pass wave-uniform predicated)
#pragma unroll
        for (int it = 0; it < 3; ++it) {
            const int e = t + it * 192;
            if (e < MAX_W * 16) {
                const int row = e >> 4, c4 = (e & 15) << 2;
                async_b128(caps_c + ((size_t)row * DDIM) + kbase + c4,
                           lds_base + (uint32_t)(((row * LSTR) + c4) << 2));
            }
        }
        // imgs: 48 rows x 16 b128, rows >= 36 clamped (values masked later)
#pragma unroll
        for (int it = 0; it < 4; ++it) {
            const int e = t + it * 192;
            const int row = e >> 4, c4 = (e & 15) << 2;
            const int grow = min(row, MAX_R - 1);
            async_b128(imgs_i + ((size_t)grow * DDIM) + kbase + c4,
                       lds_base + (uint32_t)((((MAX_W + row) * LSTR) + c4) << 2));
        }
    };

    // ---- Phase 1: WMMA sims (raw dot products, temperature applied later) ----
    stage_chunk(base0, 0);
    wait_async0();
    __syncthreads();

    const int wt = wid / 3, rt = wid % 3;           // 2 w-tiles x 3 r-tiles
    const int aoff = (wt * 16 + l15) * LSTR + kl;   // caps fragment (dwords)
    const int boff = (MAX_W + rt * 16 + l15) * LSTR + kl;

    v8f acc = {};
    for (int ch = 0; ch < NCHUNK; ++ch) {
        if (ch + 1 < NCHUNK)                         // prefetch next chunk (other buffer)
            stage_chunk((ch & 1) ? base0 : base1, (ch + 1) * KCH);
        const float* buf = (ch & 1) ? &s_stage[1][0] : &s_stage[0][0];
#pragma unroll
        for (int kk = 0; kk < KCH; kk += 4) {
            v2f a = *(const v2f*)(buf + aoff + kk);
            v2f b = *(const v2f*)(buf + boff + kk);
            acc = __builtin_amdgcn_wmma_f32_16x16x4_f32(false, a, false, b,
                                                        (short)0, acc, false, false);
        }
        wait_async0();
        __syncthreads();
    }
    {
        const int row0 = wt * 16 + ((lane & 16) ? 8 : 0);
        const int col  = rt * 16 + l15;
#pragma unroll
        for (int n = 0; n < 8; ++n)
            s_mat[(row0 + n) * SROW + col] = acc[n];
    }
    __syncthreads();

    const int img_len = img_lens[i];
    const int cap_len = cap_lens[c];
    float num = 0.0f;

    // ---- Phase 2: per-row top-5 threshold, softmax, num; attn back to LDS ----
    if (t < MAX_W) {
        const int w = t;
        float* row = &s_mat[w * SROW];
        if (w < cap_len) {
            float t0 = NEGV, t1 = NEGV, t2 = NEGV, t3 = NEGV, t4 = NEGV;
#pragma unroll
            for (int r = 0; r < MAX_R; ++r) {
                float vr = (r < img_len) ? row[r] * 10.0f : NEGV;   // 1/TEMPERATURE
                if (vr > t4) {
                    t4 = vr;
                    if (t4 > t3) { float x = t3; t3 = t4; t4 = x;
                      if (t3 > t2) { x = t2; t2 = t3; t3 = x;
                        if (t2 > t1) { x = t1; t1 = t2; t2 = x;
                          if (t1 > t0) { x = t0; t0 = t1; t1 = x; } } } }
                }
            }
            const float kth = t4, m = t0;
            float denom = 0.0f, nacc = 0.0f;
#pragma unroll
            for (int r = 0; r < MAX_R; ++r) {
                const float sr = row[r];
                const float vr = (r < img_len) ? sr * 10.0f : NEGV;
                const float er = (vr >= kth) ? __expf(vr - m) : 0.0f;
                denom += er;
                nacc  += er * sr;
            }
            const float inv = 1.0f / denom;
            num = nacc * inv;
#pragma unroll
            for (int r = 0; r < MAX_R; ++r) {
                const float sr = row[r];
                const float vr = (r < img_len) ? sr * 10.0f : NEGV;
                const float er = (vr >= kth) ? __expf(vr - m) : 0.0f;
                row[r] = er * inv;                  // attn
            }
        } else {
#pragma unroll
            for (int r = 0; r < MAX_R; ++r) row[r] = 0.0f;
        }
    }
    __syncthreads();

    // ---- Phase 3: den^2[w] = attn_w^T G_i attn_w, 6 threads per row ----
    {
        const int w = t / 6, j = t % 6;             // r = j, j+6, ..., j+30 covers 0..35
        const float* g    = G + (size_t)i * RPAD * RPAD;
        const float* arow = &s_mat[w * SROW];
        float partial = 0.0f;
#pragma unroll
        for (int mm = 0; mm < 6; ++mm) {
            const int r = j + 6 * mm;
            const float ar = arow[r];
            const float4* gr4 = (const float4*)(g + (size_t)r * RPAD);
            float acc2 = 0.0f;
#pragma unroll
            for (int q = 0; q < MAX_R / 4; ++q) {
                const float4 gv = gr4[q];
                acc2 += gv.x * arow[4 * q + 0] + gv.y * arow[4 * q + 1]
                      + gv.z * arow[4 * q + 2] + gv.w * arow[4 * q + 3];
            }
            partial += ar * acc2;
        }
        s_pd[t] = partial;
    }
    __syncthreads();

    if (t < MAX_W) {
        float den2 = 0.0f;
#pragma unroll
        for (int j = 0; j < 6; ++j) den2 += s_pd[t * 6 + j];
        const float den = sqrtf(den2) + (-1e-8f);          // EPS is negative
        const float val = (t < cap_len) ? (num / den) : -1.0f;
        out[((size_t)i * N_CAP + c) * MAX_W + t] = val;    // [n_img, n_cap, max_w]
    }
}

// ---------------------------------------------------------------------------
extern "C" void kernel_launch(void* const* d_in, const int* in_sizes, int n_in,
                              void* d_out, int out_size, void* d_ws, size_t ws_size,
                              hipStream_t stream) {
    (void)in_sizes; (void)n_in; (void)out_size; (void)ws_size;
    const float* imgs     = (const float*)d_in[0];
    const float* caps     = (const float*)d_in[1];
    const int*   img_lens = (const int*)d_in[2];
    const int*   cap_lens = (const int*)d_in[3];
    float* G   = (float*)d_ws;                 // 128*48*48 floats = 1.18 MB scratch
    float* out = (float*)d_out;

    gram_kernel<<<dim3(N_IMG), dim3(288), 0, stream>>>(imgs, G);
    vsc_main<<<dim3(N_CAP, N_IMG), dim3(192), 0, stream>>>(imgs, caps, img_lens,
                                                           cap_lens, G, out);
}